// MeshNodeBlock_70394513981948
// MI455X (gfx1250) — compile-verified
//
#include <hip/hip_runtime.h>

#define N_NODES 10000
#define N_EDGES 320000
#define D       256
#define HID     256
#define DIN     512

// LDS strides chosen so (stride mod 64) == 4 -> conflict-free ds_load_b64
#define CAT_STRIDE 516
#define H_STRIDE   260

typedef float v2f __attribute__((ext_vector_type(2)));
typedef float v8f __attribute__((ext_vector_type(8)));

// ---------------------------------------------------------------------------
// Zero the aggregation scratch (d_ws is poisoned, must re-zero every call)
// ---------------------------------------------------------------------------
__global__ void zero_kernel(float4* __restrict__ p, int n4) {
    int i = blockIdx.x * blockDim.x + threadIdx.x;
    int stride = gridDim.x * blockDim.x;
    for (; i < n4; i += stride) p[i] = make_float4(0.f, 0.f, 0.f, 0.f);
}

// ---------------------------------------------------------------------------
// segment_sum: scatter-add edge features onto destination nodes.
// float4 loads from HBM; f32 atomics land in the 10.2MB L2-resident agg buf.
// ---------------------------------------------------------------------------
__global__ void scatter_kernel(const float4* __restrict__ ef,
                               const int*    __restrict__ dst_idx,
                               float*        __restrict__ agg) {
    int i = blockIdx.x * blockDim.x + threadIdx.x;
    const int total = N_EDGES * (D / 4);          // 20.48M float4 chunks
    int stride = gridDim.x * blockDim.x;
    for (; i < total; i += stride) {
        int e = i >> 6;          // edge index   (64 float4 per edge)
        int c = i & 63;          // chunk within edge
        float4 v = ef[(size_t)e * 64 + c];
        int n = dst_idx[e];
        float* base = agg + (size_t)n * D + c * 4;
        atomicAdd(base + 0, v.x);
        atomicAdd(base + 1, v.y);
        atomicAdd(base + 2, v.z);
        atomicAdd(base + 3, v.w);
    }
}

// ---------------------------------------------------------------------------
// Fused MLP: cat -> Linear(512,256) -> SiLU -> Linear(256,256) -> LN -> +res
// One block = one 16-node tile. 8 waves; each wave owns 2 x 16-col WMMA tiles.
// f32 WMMA (16x16x4) keeps exact fp32 precision; compute is not the bottleneck.
// ---------------------------------------------------------------------------
__launch_bounds__(256, 2)
__global__ void mlp_kernel(const float* __restrict__ agg,
                           const float* __restrict__ node,
                           const float* __restrict__ W1,
                           const float* __restrict__ b1,
                           const float* __restrict__ W2,
                           const float* __restrict__ b2,
                           const float* __restrict__ ln_g,
                           const float* __restrict__ ln_b,
                           float*       __restrict__ out) {
    __shared__ float cat [16 * CAT_STRIDE];   // 16 x 512 (padded)
    __shared__ float hbuf[16 * H_STRIDE];     // 16 x 256 (padded)
    __shared__ float obuf[16 * H_STRIDE];     // 16 x 256 (padded)

    const int tile = blockIdx.x;
    const int tid  = threadIdx.x;
    const int wave = tid >> 5;
    const int lane = tid & 31;

    // ---- stage cat = [agg | node] for 16 rows into LDS (float4 vectorized)
    {
        const int nodeBase = tile * 16;
        for (int i = tid; i < 16 * 128; i += 256) {   // 128 float4 per row
            int r  = i >> 7;
            int c4 = i & 127;
            int n  = nodeBase + r;
            float4 v = (c4 < 64)
                ? ((const float4*)(agg  + (size_t)n * D))[c4]
                : ((const float4*)(node + (size_t)n * D))[c4 - 64];
            float* dst = &cat[r * CAT_STRIDE + c4 * 4];
            dst[0] = v.x; dst[1] = v.y; dst[2] = v.z; dst[3] = v.w;
        }
    }
    __syncthreads();

    // ISA A-frag layout (32-bit 16x4): lanes 0-15 -> K=k0+0/1, lanes 16-31 -> K=k0+2/3
    const int mrow  = lane & 15;
    const int khalf = (lane >> 4) * 2;
    const int mbase = (lane >> 4) * 8;       // C/D layout: VGPR r -> M = r (+8 for hi lanes)

    // ---- GEMM1: cat[16x512] @ W1[512x256] + b1, SiLU -> hbuf
    for (int t = 0; t < 2; ++t) {
        const int col = (wave * 2 + t) * 16 + (lane & 15);
        v8f acc;
        {
            float bias = b1[col];            // C(M,N) init = b1[N]
            #pragma unroll
            for (int r = 0; r < 8; ++r) acc[r] = bias;
        }
        for (int k0 = 0; k0 < DIN; k0 += 4) {
            v2f a, b;
            a.x = cat[mrow * CAT_STRIDE + k0 + khalf];
            a.y = cat[mrow * CAT_STRIDE + k0 + khalf + 1];
            b.x = W1[(size_t)(k0 + khalf)     * HID + col];
            b.y = W1[(size_t)(k0 + khalf + 1) * HID + col];
            acc = __builtin_amdgcn_wmma_f32_16x16x4_f32(
                false, a, false, b, (short)0, acc, false, false);
        }
        #pragma unroll
        for (int r = 0; r < 8; ++r) {
            float x = acc[r];
            // SiLU via hw reciprocal (v_rcp_f32, ~1ulp) -- avoids the IEEE
            // div_scale/div_fmas/div_fixup chain on the inter-GEMM path.
            float s = x * __builtin_amdgcn_rcpf(1.0f + __expf(-x));
            hbuf[(r + mbase) * H_STRIDE + col] = s;
        }
    }
    __syncthreads();

    // ---- GEMM2: h[16x256] @ W2[256x256] + b2 -> obuf
    for (int t = 0; t < 2; ++t) {
        const int col = (wave * 2 + t) * 16 + (lane & 15);
        v8f acc;
        {
            float bias = b2[col];
            #pragma unroll
            for (int r = 0; r < 8; ++r) acc[r] = bias;
        }
        for (int k0 = 0; k0 < HID; k0 += 4) {
            v2f a, b;
            a.x = hbuf[mrow * H_STRIDE + k0 + khalf];
            a.y = hbuf[mrow * H_STRIDE + k0 + khalf + 1];
            b.x = W2[(size_t)(k0 + khalf)     * D + col];
            b.y = W2[(size_t)(k0 + khalf + 1) * D + col];
            acc = __builtin_amdgcn_wmma_f32_16x16x4_f32(
                false, a, false, b, (short)0, acc, false, false);
        }
        #pragma unroll
        for (int r = 0; r < 8; ++r)
            obuf[(r + mbase) * H_STRIDE + col] = acc[r];
    }
    __syncthreads();

    // ---- LayerNorm + residual; each wave handles 2 rows via wave32 reduction
    for (int rr = 0; rr < 2; ++rr) {
        const int row = wave * 2 + rr;
        const int n   = tile * 16 + row;
        float s = 0.f, sq = 0.f;
        #pragma unroll
        for (int j = 0; j < 8; ++j) {
            float x = obuf[row * H_STRIDE + lane + 32 * j];
            s  += x;
            sq += x * x;
        }
        #pragma unroll
        for (int off = 16; off > 0; off >>= 1) {
            s  += __shfl_xor(s,  off, 32);
            sq += __shfl_xor(sq, off, 32);
        }
        const float mu  = s * (1.0f / 256.0f);
        const float var = sq * (1.0f / 256.0f) - mu * mu;
        const float rs  = rsqrtf(var + 1e-5f);
        #pragma unroll
        for (int j = 0; j < 8; ++j) {
            int c = lane + 32 * j;
            float x = obuf[row * H_STRIDE + c];
            float y = (x - mu) * rs * ln_g[c] + ln_b[c] + node[(size_t)n * D + c];
            out[(size_t)n * D + c] = y;
        }
    }
}

// ---------------------------------------------------------------------------
extern "C" void kernel_launch(void* const* d_in, const int* in_sizes, int n_in,
                              void* d_out, int out_size, void* d_ws, size_t ws_size,
                              hipStream_t stream) {
    const float* edge_feats = (const float*)d_in[0];
    const float* node_feats = (const float*)d_in[1];
    const int*   dst_idx    = (const int*)  d_in[2];
    const float* W1         = (const float*)d_in[3];
    const float* b1         = (const float*)d_in[4];
    const float* W2         = (const float*)d_in[5];
    const float* b2         = (const float*)d_in[6];
    const float* ln_g       = (const float*)d_in[7];
    const float* ln_b       = (const float*)d_in[8];

    float* out_edges = (float*)d_out;                       // output[0]: passthrough
    float* out_nodes = out_edges + (size_t)N_EDGES * D;     // output[1]: new node feats
    float* agg       = (float*)d_ws;                        // 10.24 MB scratch

    // 1) edge passthrough (pure bandwidth, D2D copy is optimal & capture-safe)
    hipMemcpyAsync(out_edges, edge_feats, (size_t)N_EDGES * D * sizeof(float),
                   hipMemcpyDeviceToDevice, stream);

    // 2) zero agg scratch
    zero_kernel<<<2500, 256, 0, stream>>>((float4*)agg, N_NODES * D / 4);

    // 3) segment_sum via f32 atomics (agg is L2-resident)
    scatter_kernel<<<40000, 256, 0, stream>>>((const float4*)edge_feats, dst_idx, agg);

    // 4) fused WMMA MLP + LayerNorm + residual (625 tiles of 16 nodes)
    mlp_kernel<<<625, 256, 0, stream>>>(agg, node_feats, W1, b1, W2, b2,
                                        ln_g, ln_b, out_nodes);
}